// DinkNet_22608707846322
// MI455X (gfx1250) — compile-verified
//
#include <hip/hip_runtime.h>
#include <hip/hip_bf16.h>

#define N_NODES 100000
#define N_IN    512
#define N_H     64
#define N_EDGES 1600000

typedef __attribute__((ext_vector_type(2))) float v2f;
typedef __attribute__((ext_vector_type(8))) float v8f;
typedef unsigned int u32x4 __attribute__((ext_vector_type(4)));
typedef int i32x4 __attribute__((ext_vector_type(4)));
typedef int i32x8 __attribute__((ext_vector_type(8)));

// ---------------------------------------------------------------------------
// GEMM: h[m][n] = sum_k x[m][k] * fcW[n][k]   via V_WMMA_F32_16X16X4_F32
// One wave -> 16 rows x 64 cols. Block = 8 waves = 128 rows.
// fc_weight (64x512 fp32 = 128KB) staged into LDS by the Tensor Data Mover
// (one tensor_load_to_lds from wave 0, s_wait_tensorcnt, then block barrier).
// ---------------------------------------------------------------------------
__launch_bounds__(256)
__global__ void gemm_fc_wmma(const float* __restrict__ x,
                             const float* __restrict__ fcW,
                             float* __restrict__ hbuf) {
  __shared__ float wlds[N_H * N_IN];          // 128 KB (<= 320KB/WGP)

  const int t = threadIdx.x;

#if __has_builtin(__builtin_amdgcn_tensor_load_to_lds)
  // ---- TDM stage: 2D tile, 8-byte elements: 64 rows x 256 cols, stride 256
  if ((t >> 5) == 0) {                        // wave 0 only (uniform branch)
    const unsigned long long ga = (unsigned long long)(uintptr_t)fcW;
    u32x4 g0;
    g0[0] = 1u;                                    // count=1, user mode
    g0[1] = (unsigned int)(uintptr_t)wlds;         // lds_addr (low 32 = LDS offset)
    g0[2] = (unsigned int)ga;                      // global_addr[31:0]
    g0[3] = ((unsigned int)(ga >> 32) & 0x1FFFFFFu) | (2u << 30); // addr[56:32], type=2
    i32x8 g1;
    g1[0] = (3 << 16);                             // wg_mask=0, data_size=3 (8B)
    g1[1] = (int)(256u << 16);                     // tensor_dim0 = 256 (bits 79:48 lo)
    g1[2] = (int)(64u  << 16);                     // dim0 hi=0 ; tensor_dim1 = 64
    g1[3] = (int)(256u << 16);                     // dim1 hi=0 ; tile_dim0 = 256
    g1[4] = 64;                                    // tile_dim1 = 64, tile_dim2 = 0
    g1[5] = 256;                                   // tensor_dim0_stride = 256
    g1[6] = 0;                                     // stride hi / dim1_stride lo
    g1[7] = 0;
    i32x4 z4 = {0, 0, 0, 0};                       // groups 2/3: dims unused
#if defined(__clang_major__) && (__clang_major__ >= 23)
    i32x8 z8 = {0, 0, 0, 0, 0, 0, 0, 0};
    __builtin_amdgcn_tensor_load_to_lds(g0, g1, z4, z4, z8, 0);
#else
    __builtin_amdgcn_tensor_load_to_lds(g0, g1, z4, z4, 0);
#endif
    __builtin_amdgcn_s_wait_tensorcnt(0);
  }
#else
  // ---- fallback: cooperative fp32 stage, 8192 float4 across 256 threads
  {
    const float4* src = (const float4*)fcW;
    float4* dst = (float4*)wlds;
    #pragma unroll
    for (int i = 0; i < (N_H * N_IN / 4) / 256; ++i)
      dst[t + i * 256] = src[t + i * 256];
  }
#endif
  __syncthreads();

  const int wave = t >> 5;
  const int lane = t & 31;
  const int r    = lane & 15;    // row-in-tile (A) / col-in-tile (B,D)
  const int hh   = lane >> 4;    // K-half selector per ISA layout
  const int m0   = (blockIdx.x * 8 + wave) * 16;
  if (m0 >= N_NODES) return;     // wave-uniform: EXEC stays all-1s for WMMA

  const float* __restrict__ arow = x + (size_t)(m0 + r) * N_IN + 2 * hh;
  // four loop-invariant B bases -> all inner-loop DS offsets are immediates
  const float* __restrict__ wb0 = wlds + (size_t)r * N_IN + 2 * hh;
  const float* __restrict__ wb1 = wb0 + 16 * N_IN;
  const float* __restrict__ wb2 = wb0 + 32 * N_IN;
  const float* __restrict__ wb3 = wb0 + 48 * N_IN;

  v8f acc0 = {}, acc1 = {}, acc2 = {}, acc3 = {};

  #pragma unroll 8
  for (int k = 0; k < N_IN; k += 4) {
    if ((k & 31) == 0)
      __builtin_prefetch(arow + k + 64, 0, 3);   // global_prefetch_b8, A stream

    v2f a  = *(const v2f*)(arow + k);
    v2f b0 = *(const v2f*)(wb0 + k);
    v2f b1 = *(const v2f*)(wb1 + k);
    v2f b2 = *(const v2f*)(wb2 + k);
    v2f b3 = *(const v2f*)(wb3 + k);

    acc0 = __builtin_amdgcn_wmma_f32_16x16x4_f32(false, a, false, b0, (short)0, acc0, false, false);
    acc1 = __builtin_amdgcn_wmma_f32_16x16x4_f32(false, a, false, b1, (short)0, acc1, false, false);
    acc2 = __builtin_amdgcn_wmma_f32_16x16x4_f32(false, a, false, b2, (short)0, acc2, false, false);
    acc3 = __builtin_amdgcn_wmma_f32_16x16x4_f32(false, a, false, b3, (short)0, acc3, false, false);
  }

  // D layout: VGPR v, lane -> row = m0 + v + 8*hh, col = tile*16 + r
  float* __restrict__ orow = hbuf + (size_t)(m0 + 8 * hh) * N_H + r;
  #pragma unroll
  for (int v = 0; v < 8; ++v) {
    orow[(size_t)v * N_H +  0] = acc0[v];
    orow[(size_t)v * N_H + 16] = acc1[v];
    orow[(size_t)v * N_H + 32] = acc2[v];
    orow[(size_t)v * N_H + 48] = acc3[v];
  }
}

// ---------------------------------------------------------------------------
// agg zero-init (graph-capture-safe, deterministic)
// ---------------------------------------------------------------------------
__global__ void zero_f32(float* __restrict__ p, int n) {
  int i = blockIdx.x * blockDim.x + threadIdx.x;
  if (i < n) p[i] = 0.0f;
}

// ---------------------------------------------------------------------------
// SpMM scatter: agg[row[e]][j] += w[e] * h[col[e]][j]
// 64 lanes per edge -> coalesced 256B gather + 256B atomic scatter.
// agg (25.6MB) lives in the 192MB L2, atomics resolve at L2 rate.
// ---------------------------------------------------------------------------
__launch_bounds__(256)
__global__ void spmm_scatter(const float* __restrict__ h,
                             const int* __restrict__ erow,
                             const int* __restrict__ ecol,
                             const float* __restrict__ ew,
                             float* __restrict__ agg) {
  const long long tid = (long long)blockIdx.x * 256 + threadIdx.x;
  int e = (int)(tid >> 6);
  const int j = (int)(tid & 63);
  if (e >= N_EDGES) return;
  e = __builtin_amdgcn_readfirstlane(e);     // wave-uniform -> scalar index loads
  const int   r  = erow[e];
  const int   c  = ecol[e];
  const float we = ew[e];
  atomicAdd(&agg[(size_t)r * N_H + j], we * h[(size_t)c * N_H + j]);
}

// ---------------------------------------------------------------------------
// wsum[k] = sum_j linW[j][k];  wsum[64] = sum_j linB[j]
// (final linear + row-sum collapses to one dot per node)
// ---------------------------------------------------------------------------
__global__ void reduce_lin(const float* __restrict__ linW,
                           const float* __restrict__ linB,
                           float* __restrict__ wsum) {
  const int k = threadIdx.x;     // 64 threads
  float s = 0.0f;
  for (int j = 0; j < N_H; ++j) s += linW[j * N_H + k];
  wsum[k] = s;
  if (k == 0) {
    float b = 0.0f;
    for (int j = 0; j < N_H; ++j) b += linB[j];
    wsum[N_H] = b;
  }
}

// ---------------------------------------------------------------------------
// z[n] = dot(prelu(agg[n]+bias), wsum) + bsum ; one wave32 per node
// ---------------------------------------------------------------------------
__launch_bounds__(256)
__global__ void finalize(const float* __restrict__ agg,
                         const float* __restrict__ gbias,
                         const float* __restrict__ prelu_a,
                         const float* __restrict__ wsum,
                         float* __restrict__ z) {
  const int node = (blockIdx.x * blockDim.x + threadIdx.x) >> 5;
  const int lane = threadIdx.x & 31;
  if (node >= N_NODES) return;

  const float al   = prelu_a[0];
  const float bsum = wsum[N_H];

  const float2 v  = *(const float2*)(agg   + (size_t)node * N_H + 2 * lane);
  const float2 b  = *(const float2*)(gbias + 2 * lane);
  const float2 wv = *(const float2*)(wsum  + 2 * lane);

  float t0 = v.x + b.x; t0 = (t0 >= 0.0f) ? t0 : al * t0;
  float t1 = v.y + b.y; t1 = (t1 >= 0.0f) ? t1 : al * t1;
  float p = t0 * wv.x + t1 * wv.y;

  #pragma unroll
  for (int off = 16; off > 0; off >>= 1)
    p += __shfl_down(p, off, 32);

  if (lane == 0) z[node] = p + bsum;
}

// ---------------------------------------------------------------------------
extern "C" void kernel_launch(void* const* d_in, const int* in_sizes, int n_in,
                              void* d_out, int out_size, void* d_ws, size_t ws_size,
                              hipStream_t stream) {
  const float* x1   = (const float*)d_in[0];
  const float* x2   = (const float*)d_in[1];
  const int*   erow = (const int*)  d_in[2];
  const int*   ecol = (const int*)  d_in[3];
  const float* ew   = (const float*)d_in[4];
  const float* fcW  = (const float*)d_in[5];
  const float* gb   = (const float*)d_in[6];
  const float* pa   = (const float*)d_in[7];
  const float* linW = (const float*)d_in[8];
  const float* linB = (const float*)d_in[9];
  float* out = (float*)d_out;

  float* hbuf = (float*)d_ws;                         // 100000*64 f32 = 25.6MB
  float* agg  = hbuf + (size_t)N_NODES * N_H;         // 25.6MB
  float* wsum = agg  + (size_t)N_NODES * N_H;         // 65 f32

  const int gemmBlocks = (N_NODES / 16 + 7) / 8;      // 782
  const int zeroBlocks = (N_NODES * N_H) / 256;       // 25000
  const int spmmBlocks = ((long long)N_EDGES * 64) / 256;  // 400000
  const int finBlocks  = (N_NODES * 32) / 256;        // 12500

  reduce_lin<<<1, 64, 0, stream>>>(linW, linB, wsum);

  const float* xs[2] = {x1, x2};
  for (int pass = 0; pass < 2; ++pass) {
    gemm_fc_wmma<<<gemmBlocks, 256, 0, stream>>>(xs[pass], fcW, hbuf);
    zero_f32<<<zeroBlocks, 256, 0, stream>>>(agg, N_NODES * N_H);
    spmm_scatter<<<spmmBlocks, 256, 0, stream>>>(hbuf, erow, ecol, ew, agg);
    finalize<<<finBlocks, 256, 0, stream>>>(agg, gb, pa, wsum, out + (size_t)pass * N_NODES);
  }
}